// GINBlock_22754736734325
// MI455X (gfx1250) — compile-verified
//
#include <hip/hip_runtime.h>
#include <math.h>

#define FD 128            // feature dim D
#define BN_EPS 1e-5f

typedef float f32x2 __attribute__((ext_vector_type(2)));
typedef float f32x4 __attribute__((ext_vector_type(4)));
typedef float f32x8 __attribute__((ext_vector_type(8)));

// ---------------- zero the aggregation buffer ----------------
__global__ void zero_kernel(float* __restrict__ p, int n4) {
    int i = blockIdx.x * blockDim.x + threadIdx.x;
    if (i < n4) ((f32x4*)p)[i] = (f32x4)0.0f;
}

// ---------------- edge scatter-add: agg[dst] += x[src] ----------------
// one thread per (edge, 4-feature chunk); D/4 == 32 chunks per edge
__global__ void scatter_add_kernel(const float* __restrict__ x,
                                   const int* __restrict__ src,
                                   const int* __restrict__ dst,
                                   float* __restrict__ agg, int E) {
    int t = blockIdx.x * blockDim.x + threadIdx.x;
    int total = E * (FD / 4);
    if (t >= total) return;
    int e = t >> 5;             // FD/4 == 32
    int q = (t & 31) << 2;
    int s = src[e];
    int d = dst[e];
    f32x4 v = *(const f32x4*)(x + (size_t)s * FD + q);
    float* ap = agg + (size_t)d * FD + q;
    atomicAdd(ap + 0, v.x);
    atomicAdd(ap + 1, v.y);
    atomicAdd(ap + 2, v.z);
    atomicAdd(ap + 3, v.w);
}

// ---------------- fused (1+eps)*x + agg, then @ W^T + b via WMMA ----------------
// grid.x = N/16 (M tiles). 256 threads = 8 waves; wave w handles column tile w.
// W (row-major [out][in], 64KB) staged in LDS once per block.
__global__ void __launch_bounds__(256)
gin_gemm_wmma(const float* __restrict__ x, const float* __restrict__ agg,
              const float* __restrict__ W, const float* __restrict__ bias,
              const float* __restrict__ gin_eps, float* __restrict__ h) {
    __shared__ float ldsW[FD * FD];     // 64 KB of the WGP's 320 KB LDS

    // cooperative 128-bit copies of W into LDS
    {
        const f32x4* Wv = (const f32x4*)W;
        f32x4*       Lv = (f32x4*)ldsW;
        for (int i = threadIdx.x; i < FD * FD / 4; i += 256) Lv[i] = Wv[i];
    }
    __syncthreads();

    const float ep1  = 1.0f + gin_eps[0];
    const int   lane = threadIdx.x & 31;
    const int   wv   = threadIdx.x >> 5;     // 0..7 -> N tile
    const int   lh   = lane & 15;
    const int   hi   = (lane >> 4) & 1;      // lanes 16..31 carry K+2/K+3
    const int   mt   = blockIdx.x;           // M tile

    const int rowA = mt * 16 + lh;           // A-matrix row for this lane
    const int col  = wv * 16 + lh;           // B/D column for this lane
    const float* xr = x   + (size_t)rowA * FD;
    const float* ar = agg + (size_t)rowA * FD;
    const float* wc = ldsW + col * FD;       // row `col` of W == column of W^T

    f32x8 c = (f32x8)0.0f;
#pragma unroll
    for (int k0 = 0; k0 < FD; k0 += 4) {
        const int kk = k0 + 2 * hi;
        // A fragment: a = (1+eps)*x + agg, ISA 16x4 f32 layout (2 VGPRs)
        f32x2 xv = *(const f32x2*)(xr + kk);
        f32x2 av = *(const f32x2*)(ar + kk);
        f32x2 a;
        a.x = ep1 * xv.x + av.x;
        a.y = ep1 * xv.y + av.y;
        // B fragment: W^T[kk..kk+1][col] == contiguous W[col][kk..kk+1] from LDS
        f32x2 b = *(const f32x2*)(wc + kk);
        c = __builtin_amdgcn_wmma_f32_16x16x4_f32(
                /*neg_a=*/false, a, /*neg_b=*/false, b,
                /*c_mod=*/(short)0, c, /*reuse_a=*/false, /*reuse_b=*/false);
    }

    const float bb = bias[col];
    // D-matrix layout: VGPR r holds M=r (lanes 0-15) and M=r+8 (lanes 16-31)
#pragma unroll
    for (int r = 0; r < 8; ++r) {
        int row = mt * 16 + r + hi * 8;
        h[(size_t)row * FD + col] = c[r] + bb;
    }
}

// ---------------- per-feature batch stats: mean + 1/sqrt(var+eps) ----------------
__global__ void bn_stats_kernel(const float* __restrict__ h,
                                float* __restrict__ mean,
                                float* __restrict__ invstd, int N) {
    __shared__ float ssum[256];
    __shared__ float ssq[256];
    const int f = blockIdx.x;
    float s = 0.0f, q = 0.0f;
    for (int i = threadIdx.x; i < N; i += 256) {
        float v = h[(size_t)i * FD + f];
        s += v;
        q += v * v;
    }
    ssum[threadIdx.x] = s;
    ssq[threadIdx.x]  = q;
    __syncthreads();
    for (int off = 128; off > 0; off >>= 1) {
        if (threadIdx.x < off) {
            ssum[threadIdx.x] += ssum[threadIdx.x + off];
            ssq[threadIdx.x]  += ssq[threadIdx.x + off];
        }
        __syncthreads();
    }
    if (threadIdx.x == 0) {
        float m   = ssum[0] / (float)N;
        float var = ssq[0] / (float)N - m * m;   // biased variance
        mean[f]   = m;
        invstd[f] = rsqrtf(var + BN_EPS);
    }
}

// ---------------- BN affine + ReLU + residual, in place on d_out ----------------
__global__ void bn_relu_res_kernel(float* __restrict__ h, const float* __restrict__ x,
                                   const float* __restrict__ gamma,
                                   const float* __restrict__ beta,
                                   const float* __restrict__ mean,
                                   const float* __restrict__ invstd, int total4) {
    int i = blockIdx.x * blockDim.x + threadIdx.x;
    if (i >= total4) return;
    int f0 = (i * 4) & (FD - 1);
    f32x4 hv = ((const f32x4*)h)[i];
    f32x4 xv = ((const f32x4*)x)[i];
    f32x4 g  = *(const f32x4*)(gamma  + f0);
    f32x4 bt = *(const f32x4*)(beta   + f0);
    f32x4 mn = *(const f32x4*)(mean   + f0);
    f32x4 is = *(const f32x4*)(invstd + f0);
    f32x4 o;
#pragma unroll
    for (int j = 0; j < 4; ++j) {
        float v = g[j] * (hv[j] - mn[j]) * is[j] + bt[j];
        v = v > 0.0f ? v : 0.0f;
        o[j] = v + xv[j];
    }
    ((f32x4*)h)[i] = o;
}

extern "C" void kernel_launch(void* const* d_in, const int* in_sizes, int n_in,
                              void* d_out, int out_size, void* d_ws, size_t ws_size,
                              hipStream_t stream) {
    const float* x     = (const float*)d_in[0];
    const int*   ei    = (const int*)d_in[1];
    const float* W     = (const float*)d_in[2];
    const float* bias  = (const float*)d_in[3];
    const float* gamma = (const float*)d_in[4];
    const float* beta  = (const float*)d_in[5];
    const float* geps  = (const float*)d_in[6];

    const int N = in_sizes[0] / FD;
    const int E = in_sizes[1] / 2;
    const int* src = ei;         // edge_index[0]
    const int* dst = ei + E;     // edge_index[1]

    // workspace: agg [N*D] | mean [D] | invstd [D]   (~20.5 MB)
    float* agg    = (float*)d_ws;
    float* mean   = agg + (size_t)N * FD;
    float* invstd = mean + FD;
    float* h      = (float*)d_out;   // GEMM output lives in d_out, finalized in place

    const int n4 = N * FD / 4;
    zero_kernel<<<(n4 + 255) / 256, 256, 0, stream>>>(agg, n4);

    const int st = E * (FD / 4);
    scatter_add_kernel<<<(st + 255) / 256, 256, 0, stream>>>(x, src, dst, agg, E);

    gin_gemm_wmma<<<N / 16, 256, 0, stream>>>(x, agg, W, bias, geps, h);

    bn_stats_kernel<<<FD, 256, 0, stream>>>(h, mean, invstd, N);

    bn_relu_res_kernel<<<(n4 + 255) / 256, 256, 0, stream>>>(h, x, gamma, beta,
                                                             mean, invstd, n4);
}